// SpikeFP32Sigmoid_29875792511497
// MI455X (gfx1250) — compile-verified
//
#include <hip/hip_runtime.h>
#include <hip/hip_bf16.h>
#include <stdint.h>

// -----------------------------------------------------------------------------
// Bit-serial fp32 sigmoid on MI455X (gfx1250).
// Bandwidth-bound: 1 GiB traffic @ 23.3 TB/s => ~46us floor. Data movement via
// the CDNA5 Tensor Data Mover (one 32KB DMA per block per direction), compute
// out of LDS in exact fp32 (output must be the exact IEEE bit pattern).
// -----------------------------------------------------------------------------

typedef uint32_t v4u __attribute__((ext_vector_type(4)));
typedef int      v4i __attribute__((ext_vector_type(4)));
typedef int      v8i __attribute__((ext_vector_type(8)));

#define TILE_ELEMS  256                   // fp32 values per block
#define TILE_FLOATS (TILE_ELEMS * 32)     // 8192 pulse floats = 32 KB

// Build the Tensor DMA Descriptor (D#) for a contiguous run of `nfloats`
// fp32 elements at `gaddr` <-> LDS byte offset `lds_addr`.
// Layout per CDNA5 ISA 8.3-8.6: 1-row 2D tensor, data_size=4B, no padding,
// no multicast (wg_mask=0, we are not in a cluster), no iterate/gather.
__device__ __forceinline__ void tdm_build(v4u& g0, v8i& g1,
                                          uint32_t lds_addr, uint64_t gaddr,
                                          uint32_t nfloats) {
  g0.x = 1u;                                         // count=1 (valid), user mode
  g0.y = lds_addr;                                   // LDS byte address
  g0.z = (uint32_t)gaddr;                            // global_addr[31:0]
  g0.w = ((uint32_t)(gaddr >> 32) & 0x01FFFFFFu)     // global_addr[56:32]
         | (2u << 30);                               // type=2 ("image")

  g1[0] = (int)(2u << 16);                           // wg_mask=0, data_size=2 (4B)
  g1[1] = (int)((nfloats & 0xFFFFu) << 16);          // tensor_dim0[15:0]
  g1[2] = (int)(((nfloats >> 16) & 0xFFFFu)          // tensor_dim0[31:16]
         | (1u << 16));                              // tensor_dim1 = 1
  g1[3] = (int)((nfloats & 0xFFFFu) << 16);          // tile_dim0 = nfloats (<=8192)
  g1[4] = 1;                                         // tile_dim1=1, tile_dim2=0
  g1[5] = (int)nfloats;                              // tensor_dim0_stride
  g1[6] = 0;
  g1[7] = 0;
}

__device__ __forceinline__ void tdm_load(uint32_t lds_addr, uint64_t gaddr,
                                         uint32_t nfloats) {
  v4u g0; v8i g1; tdm_build(g0, g1, lds_addr, gaddr, nfloats);
  v4i gz = (v4i)(0);
#if __clang_major__ >= 23
  v8i z8 = (v8i)(0);
  __builtin_amdgcn_tensor_load_to_lds(g0, g1, gz, gz, z8, 0);
#else
  __builtin_amdgcn_tensor_load_to_lds(g0, g1, gz, gz, 0);
#endif
}

__device__ __forceinline__ void tdm_store(uint32_t lds_addr, uint64_t gaddr,
                                          uint32_t nfloats) {
  v4u g0; v8i g1; tdm_build(g0, g1, lds_addr, gaddr, nfloats);
  v4i gz = (v4i)(0);
#if __clang_major__ >= 23
  v8i z8 = (v8i)(0);
  __builtin_amdgcn_tensor_store_from_lds(g0, g1, gz, gz, z8, 0);
#else
  __builtin_amdgcn_tensor_store_from_lds(g0, g1, gz, gz, 0);
#endif
}

__global__ void __launch_bounds__(TILE_ELEMS)
spike_fp32_sigmoid_kernel(const float* __restrict__ x,
                          float* __restrict__ out,
                          unsigned total_floats) {
  __shared__ __align__(16) float tin[TILE_FLOATS];
  __shared__ __align__(16) float tout[TILE_FLOATS];

  const unsigned base = blockIdx.x * (unsigned)TILE_FLOATS;
  const unsigned rem  = total_floats - base;
  const unsigned nflt = rem < (unsigned)TILE_FLOATS ? rem : (unsigned)TILE_FLOATS;
  const unsigned nelem = nflt >> 5;

  // --- Stage in-tile via TDM (one DMA per block; EXEC-independent, so gate
  // --- to a single wave). TENSORcnt tracks completion.
  if (threadIdx.x == 0) {
    tdm_load((uint32_t)(uintptr_t)(void*)tin,
             (uint64_t)(uintptr_t)(x + base), nflt);
    __builtin_amdgcn_s_wait_tensorcnt(0);
  }
  __syncthreads();

  const unsigned t = threadIdx.x;
  if (t < nelem) {
    // Decode: pulses are exact 0.0/1.0, MSB-first IEEE-754. Reconstruct the
    // bit pattern directly (bit-exact vs the reference weighted-sum decode).
    const float4* src = (const float4*)(tin + t * 32u);
    uint32_t u = 0u;
#pragma unroll
    for (int k = 0; k < 8; ++k) {
      float4 p = src[k];
      u = (u << 1) | (uint32_t)(p.x != 0.0f);
      u = (u << 1) | (uint32_t)(p.y != 0.0f);
      u = (u << 1) | (uint32_t)(p.z != 0.0f);
      u = (u << 1) | (uint32_t)(p.w != 0.0f);
    }
    // NOT gate on sign pulse == sign-bit flip: v = -x
    float v = __uint_as_float(u ^ 0x80000000u);
    // sigmoid(x) = 1 / (1 + exp(-x)), exact fp32 add + IEEE divide
    float r = 1.0f / (1.0f + expf(v));

    // Encode: exact IEEE bit pattern of r, MSB first, as 0.0/1.0 floats.
    uint32_t b = __float_as_uint(r);
    float4* dst = (float4*)(tout + t * 32u);
#pragma unroll
    for (int k = 0; k < 8; ++k) {
      const int j = k * 4;
      float4 q;
      q.x = (float)((b >> (31 - j)) & 1u);
      q.y = (float)((b >> (30 - j)) & 1u);
      q.z = (float)((b >> (29 - j)) & 1u);
      q.w = (float)((b >> (28 - j)) & 1u);
      dst[k] = q;
    }
  }
  __syncthreads();

  // --- Drain out-tile via TDM. S_ENDPGM implies wait-idle, but wait
  // --- explicitly so the issuing wave owns the LDS until the DMA is done.
  if (threadIdx.x == 0) {
    tdm_store((uint32_t)(uintptr_t)(void*)tout,
              (uint64_t)(uintptr_t)(out + base), nflt);
    __builtin_amdgcn_s_wait_tensorcnt(0);
  }
}

extern "C" void kernel_launch(void* const* d_in, const int* in_sizes, int n_in,
                              void* d_out, int out_size, void* d_ws, size_t ws_size,
                              hipStream_t stream) {
  (void)n_in; (void)out_size; (void)d_ws; (void)ws_size;
  const float* x  = (const float*)d_in[0];
  float* out      = (float*)d_out;
  const unsigned total  = (unsigned)in_sizes[0];                 // 4194304*32
  const unsigned blocks = (total + TILE_FLOATS - 1) / TILE_FLOATS; // 16384
  spike_fp32_sigmoid_kernel<<<blocks, TILE_ELEMS, 0, stream>>>(x, out, total);
}